// LookaheadModel_35270271435159
// MI455X (gfx1250) — compile-verified
//
#include <hip/hip_runtime.h>
#include <math.h>

#define Bz 8
#define Tz 4096
#define Dz 512
#define Vz 50257
#define SLOTSz 256
#define Kwin 8

typedef __attribute__((ext_vector_type(2))) float v2f;
typedef __attribute__((ext_vector_type(8))) float v8f;

__device__ __forceinline__ v8f wmma_f32(v2f a, v2f b, v8f c) {
  // 8 args: (neg_a, A, neg_b, B, c_mod, C, reuse_a, reuse_b)
  return __builtin_amdgcn_wmma_f32_16x16x4_f32(false, a, false, b, (short)0, c,
                                               false, false);
}

// Async global -> LDS copy (CDNA5, ASYNCcnt-tracked). lds is a byte offset in
// the workgroup LDS window (low 32 bits of the generic address).
__device__ __forceinline__ void async_load_b128(const float* lds,
                                                const float* gaddr) {
  unsigned l = (unsigned)(uintptr_t)lds;
  asm volatile("global_load_async_to_lds_b128 %0, %1, off" ::"v"(l), "v"(gaddr)
               : "memory");
}
__device__ __forceinline__ void wait_async0() {
  asm volatile("s_wait_asynccnt 0x0" ::: "memory");
}

// ---------------------------------------------------------------- gather h0
__global__ __launch_bounds__(128) void gather_kernel(
    const int* __restrict__ seq, const float* __restrict__ embed,
    float* __restrict__ h0) {
  int row = blockIdx.x;  // 0 .. B*T-1
  int tok = seq[row];
  const float4* src = (const float4*)(embed + (size_t)tok * Dz);
  float4* dst = (float4*)(h0 + (size_t)row * Dz);
  dst[threadIdx.x] = src[threadIdx.x];  // 128 * float4 = 512 floats
}

// ---------------------------------------------------------------- WMMA GEMM
// C[M,N] = act(A[M,K] @ B[K,N] + bias[N] (+ resid[M,N]))
// 128x64 block tile, 8 waves in 4x2 grid, 32x32 per wave (2x2 WMMA tiles).
// A tile double-buffered via async global->LDS; B tile transposed into LDS so
// fragments are contiguous ds_load_b64.
#define GBM 128
#define GBN 64
#define GKS 16
#define AST 20  // A LDS row stride (pad: conflict-free, 8B aligned)
#define BST 18  // B^T LDS row stride (pad: conflict-free, 8B aligned)

template <bool RELU, bool RESID>
__global__ __launch_bounds__(256) void gemm_wmma_kernel(
    const float* __restrict__ A, const float* __restrict__ Bm,
    const float* __restrict__ bias, const float* resid, float* C, int M, int N,
    int K) {
  __shared__ __align__(16) float As[2][GBM * AST];
  __shared__ __align__(16) float Bs[2][GBN * BST];  // transposed: [n][k]

  const int tid = threadIdx.x;
  const int lane = tid & 31;
  const int wave = tid >> 5;  // 0..7
  const int wm = wave >> 1;   // 0..3
  const int wn = wave & 1;    // 0..1
  const int mb = blockIdx.y * GBM;
  const int nb = blockIdx.x * GBN;

  const v8f vzero = {0.f, 0.f, 0.f, 0.f, 0.f, 0.f, 0.f, 0.f};
  v8f acc[2][2];
  acc[0][0] = vzero; acc[0][1] = vzero;
  acc[1][0] = vzero; acc[1][1] = vzero;

  const int l15 = lane & 15;
  const int khalf = (lane >> 4) << 1;  // lanes 16-31 hold K+2,K+3 halves

  // Per-thread A async-load coords: 128x16 tile = 512 float4, 2 per thread.
  const int ar0 = tid >> 2, ac0 = (tid & 3) << 2;
  const int ar1 = (tid + 256) >> 2, ac1 = ((tid + 256) & 3) << 2;
  // Per-thread B load coords: 16x64 tile = 1 float4 per thread, transposed in.
  const int brow = tid >> 4;         // k within tile
  const int bcol = (tid & 15) << 2;  // n offset

  // ---- prologue: tile 0
  async_load_b128(&As[0][ar0 * AST + ac0], A + (size_t)(mb + ar0) * K + ac0);
  async_load_b128(&As[0][ar1 * AST + ac1], A + (size_t)(mb + ar1) * K + ac1);
  {
    float4 v = *(const float4*)(Bm + (size_t)brow * N + nb + bcol);
    float* p = &Bs[0][0];
    p[(bcol + 0) * BST + brow] = v.x;
    p[(bcol + 1) * BST + brow] = v.y;
    p[(bcol + 2) * BST + brow] = v.z;
    p[(bcol + 3) * BST + brow] = v.w;
  }
  wait_async0();
  __syncthreads();

  int cur = 0;
  for (int kb = 0; kb < K; kb += GKS) {
    const int nxt = cur ^ 1;
    const bool has_next = (kb + GKS) < K;
    float4 bv;
    if (has_next) {
      const int kn = kb + GKS;
      async_load_b128(&As[nxt][ar0 * AST + ac0],
                      A + (size_t)(mb + ar0) * K + kn + ac0);
      async_load_b128(&As[nxt][ar1 * AST + ac1],
                      A + (size_t)(mb + ar1) * K + kn + ac1);
      bv = *(const float4*)(Bm + (size_t)(kn + brow) * N + nb + bcol);
    }

    const float* Ab = &As[cur][0];
    const float* Bb = &Bs[cur][0];
#pragma unroll
    for (int kk = 0; kk < GKS; kk += 4) {
      int ko = kk + khalf;
      v2f a0, a1, b0, b1;
      const float* ap0 = Ab + (wm * 32 + l15) * AST + ko;
      a0.x = ap0[0]; a0.y = ap0[1];
      const float* ap1 = Ab + (wm * 32 + 16 + l15) * AST + ko;
      a1.x = ap1[0]; a1.y = ap1[1];
      const float* bp0 = Bb + (wn * 32 + l15) * BST + ko;
      b0.x = bp0[0]; b0.y = bp0[1];
      const float* bp1 = Bb + (wn * 32 + 16 + l15) * BST + ko;
      b1.x = bp1[0]; b1.y = bp1[1];
      acc[0][0] = wmma_f32(a0, b0, acc[0][0]);
      acc[0][1] = wmma_f32(a0, b1, acc[0][1]);
      acc[1][0] = wmma_f32(a1, b0, acc[1][0]);
      acc[1][1] = wmma_f32(a1, b1, acc[1][1]);
    }

    if (has_next) {  // store B^T for next tile after compute (latency hidden)
      float* p = &Bs[nxt][0];
      p[(bcol + 0) * BST + brow] = bv.x;
      p[(bcol + 1) * BST + brow] = bv.y;
      p[(bcol + 2) * BST + brow] = bv.z;
      p[(bcol + 3) * BST + brow] = bv.w;
    }
    wait_async0();
    __syncthreads();
    cur = nxt;
  }

  // Epilogue: D layout = elem e -> row e (+8 for lanes>=16), col = lane&15
  const int rbase = mb + wm * 32 + ((lane >> 4) << 3);
  const int cbase = nb + wn * 32 + l15;
#pragma unroll
  for (int i = 0; i < 2; i++) {
#pragma unroll
    for (int j = 0; j < 2; j++) {
      int col = cbase + j * 16;
      float bvv = bias[col];
#pragma unroll
      for (int e = 0; e < 8; e++) {
        int row = rbase + i * 16 + e;
        size_t off = (size_t)row * N + col;
        float v = acc[i][j][e] + bvv;
        if (RESID) v += resid[off];
        if (RELU) v = fmaxf(v, 0.f);
        C[off] = v;
      }
    }
  }
}

// ---------------------------------------------------------------- reductions
__device__ __forceinline__ float block_reduce_sum(float v, float* red, int tid) {
  red[tid] = v;
  __syncthreads();
  for (int s = 128; s > 0; s >>= 1) {
    if (tid < s) red[tid] += red[tid + s];
    __syncthreads();
  }
  float r = red[0];
  __syncthreads();
  return r;
}

// ------------------------------------------- fused LayerNorm + gate row-dots
__global__ __launch_bounds__(256) void ln_rowdot_kernel(
    float* hbuf, const float* __restrict__ lng, const float* __restrict__ lnb,
    const float* __restrict__ gw, float* __restrict__ s_lo,
    float* __restrict__ s_hi) {
  __shared__ float red[256];
  const int row = blockIdx.x;
  const int tid = threadIdx.x;
  float* x = hbuf + (size_t)row * Dz;
  float x0 = x[tid], x1 = x[tid + 256];
  float mean = block_reduce_sum(x0 + x1, red, tid) * (1.f / (float)Dz);
  float d0 = x0 - mean, d1 = x1 - mean;
  float var = block_reduce_sum(d0 * d0 + d1 * d1, red, tid) * (1.f / (float)Dz);
  float inv = rsqrtf(var + 1e-5f);
  float h0v = d0 * inv * lng[tid] + lnb[tid];
  float h1v = d1 * inv * lng[tid + 256] + lnb[tid + 256];
  x[tid] = h0v;
  x[tid + 256] = h1v;
  float lo = block_reduce_sum(h0v * gw[tid] + h1v * gw[tid + 256], red, tid);
  float hi =
      block_reduce_sum(h0v * gw[Dz + tid] + h1v * gw[Dz + tid + 256], red, tid);
  if (tid == 0) {
    s_lo[row] = lo;
    s_hi[row] = hi;
  }
}

// ---------------------------------------------------------------- gate
__global__ void gate_kernel(const float* __restrict__ s_lo,
                            const float* __restrict__ s_hi,
                            const float* __restrict__ gate_b,
                            float* __restrict__ gate) {
  int i = blockIdx.x * blockDim.x + threadIdx.x;
  if (i >= Bz * Tz) return;
  int t = i & (Tz - 1);
  int base = i - t;
  int start = t + 1;
  int end = t + 1 + Kwin;
  if (end > Tz) end = Tz;
  if (start > Tz) start = Tz;
  int cnt = end - start;
  float win = 0.f;
  for (int j = start; j < end; j++) win += s_hi[base + j];
  float xg = s_lo[i] + (cnt > 0 ? win / (float)cnt : 0.f) + gate_b[0];
  gate[i] = 1.f / (1.f + expf(-xg));
}

// ------------------------------------- exact top-256 by rank (deterministic)
__global__ __launch_bounds__(1024) void topk_kernel(
    const float* __restrict__ gate, int* __restrict__ mem_idx) {
  __shared__ float g[Tz];
  const int b = blockIdx.x;
  const int tid = threadIdx.x;
  const float* gb = gate + b * Tz;
  for (int i = tid; i < Tz; i += 1024) g[i] = gb[i];
  __syncthreads();
  int i0 = tid, i1 = tid + 1024, i2 = tid + 2048, i3 = tid + 3072;
  float g0 = g[i0], g1 = g[i1], g2 = g[i2], g3 = g[i3];
  int r0 = 0, r1 = 0, r2 = 0, r3 = 0;
  for (int j = 0; j < Tz; j++) {
    float gj = g[j];
    r0 += (gj > g0) || (gj == g0 && j < i0);
    r1 += (gj > g1) || (gj == g1 && j < i1);
    r2 += (gj > g2) || (gj == g2 && j < i2);
    r3 += (gj > g3) || (gj == g3 && j < i3);
  }
  if (r0 < SLOTSz) mem_idx[b * SLOTSz + r0] = i0;
  if (r1 < SLOTSz) mem_idx[b * SLOTSz + r1] = i1;
  if (r2 < SLOTSz) mem_idx[b * SLOTSz + r2] = i2;
  if (r3 < SLOTSz) mem_idx[b * SLOTSz + r3] = i3;
}

// -------------------------------------------------- q = h_last @ q_W + q_b
__global__ __launch_bounds__(256) void qproj_kernel(
    const float* __restrict__ h, const float* __restrict__ qW,
    const float* __restrict__ qb, float* __restrict__ q) {
  __shared__ float hl[Dz];
  const int b = blockIdx.x;
  const int tid = threadIdx.x;
  const float* hlast = h + ((size_t)b * Tz + (Tz - 1)) * Dz;
  hl[tid] = hlast[tid];
  hl[tid + 256] = hlast[tid + 256];
  __syncthreads();
  for (int n = tid; n < Dz; n += 256) {
    float acc = qb[n];
    for (int d = 0; d < Dz; d++) acc += hl[d] * qW[d * Dz + n];
    q[b * Dz + n] = acc;
  }
}

// ------------------------------------------------- attention read head
__global__ __launch_bounds__(256) void readhead_kernel(
    const float* __restrict__ h, const int* __restrict__ mem_idx,
    const float* __restrict__ q, float* __restrict__ ctx) {
  __shared__ float qs[Dz];
  __shared__ int idx[SLOTSz];
  __shared__ float attn[SLOTSz];
  __shared__ float red[256];
  const int b = blockIdx.x;
  const int tid = threadIdx.x;
  qs[tid] = q[b * Dz + tid];
  qs[tid + 256] = q[b * Dz + tid + 256];
  idx[tid] = mem_idx[b * SLOTSz + tid];
  __syncthreads();
  const float* hr = h + ((size_t)b * Tz + idx[tid]) * Dz;
  float s = 0.f;
  for (int d = 0; d < Dz; d++) s += hr[d] * qs[d];
  red[tid] = s;
  __syncthreads();
  for (int st = 128; st > 0; st >>= 1) {
    if (tid < st) red[tid] = fmaxf(red[tid], red[tid + st]);
    __syncthreads();
  }
  float mx = red[0];
  __syncthreads();
  float e = expf(s - mx);
  float denom = block_reduce_sum(e, red, tid);
  attn[tid] = e / denom;
  __syncthreads();
  for (int d = tid; d < Dz; d += 256) {
    float acc = 0.f;
    for (int m = 0; m < SLOTSz; m++)
      acc += attn[m] * h[((size_t)b * Tz + idx[m]) * Dz + d];
    ctx[b * Dz + d] = acc;
  }
}

// --------------------------------------------- out = ctx @ out_W + out_b
__global__ __launch_bounds__(256) void outproj_kernel(
    const float* __restrict__ ctx, const float* __restrict__ outW,
    const float* __restrict__ outb, float* __restrict__ out) {
  __shared__ float c[Bz * Dz];
  const int tid = threadIdx.x;
  for (int i = tid; i < Bz * Dz; i += 256) c[i] = ctx[i];
  __syncthreads();
  int v = blockIdx.x * 256 + tid;
  if (v >= Vz) return;
  float acc[Bz];
  float bb = outb[v];
#pragma unroll
  for (int b = 0; b < Bz; b++) acc[b] = bb;
  for (int d0 = 0; d0 < Dz; d0 += 16) {
    // stream out_W ahead of the FMA loop (speculative prefetch; OOB is safe)
    __builtin_prefetch(outW + (size_t)(d0 + 48) * Vz + v, 0, 0);
#pragma unroll
    for (int dd = 0; dd < 16; dd++) {
      int d = d0 + dd;
      float w = outW[(size_t)d * Vz + v];
#pragma unroll
      for (int b = 0; b < Bz; b++) acc[b] += c[b * Dz + d] * w;
    }
  }
#pragma unroll
  for (int b = 0; b < Bz; b++) out[(size_t)b * Vz + v] = acc[b];
}

// ---------------------------------------------------------------- launch
extern "C" void kernel_launch(void* const* d_in, const int* in_sizes, int n_in,
                              void* d_out, int out_size, void* d_ws,
                              size_t ws_size, hipStream_t stream) {
  const int* seq = (const int*)d_in[0];
  const float* embed = (const float*)d_in[1];
  const float* W1 = (const float*)d_in[2];
  const float* b1 = (const float*)d_in[3];
  const float* W2 = (const float*)d_in[4];
  const float* b2 = (const float*)d_in[5];
  const float* ln_g = (const float*)d_in[6];
  const float* ln_b = (const float*)d_in[7];
  const float* gate_W = (const float*)d_in[8];
  const float* gate_b = (const float*)d_in[9];
  const float* q_W = (const float*)d_in[10];
  const float* q_b = (const float*)d_in[11];
  const float* out_W = (const float*)d_in[12];
  const float* out_b = (const float*)d_in[13];

  char* ws = (char*)d_ws;
  size_t off = 0;
  auto alloc = [&](size_t bytes) -> void* {
    void* p = ws + off;
    off += (bytes + 255) & ~(size_t)255;
    return p;
  };
  const size_t MT = (size_t)Bz * Tz;            // 32768 rows
  float* h0 = (float*)alloc(MT * Dz * 4);       // h0 -> (h0+ff+b2) -> h
  float* ff1 = (float*)alloc(MT * 2 * Dz * 4);  // GEMM1 output
  float* s_lo = (float*)alloc(MT * 4);
  float* s_hi = (float*)alloc(MT * 4);
  float* gate = (float*)alloc(MT * 4);
  int* mem_idx = (int*)alloc((size_t)Bz * SLOTSz * 4);
  float* q = (float*)alloc((size_t)Bz * Dz * 4);
  float* ctx = (float*)alloc((size_t)Bz * Dz * 4);

  // 1) h0 = embed[seq]
  gather_kernel<<<(int)MT, 128, 0, stream>>>(seq, embed, h0);

  // 2) ff1 = relu(h0 @ W1 + b1)     M=32768, N=1024, K=512
  gemm_wmma_kernel<true, false>
      <<<dim3(2 * Dz / GBN, (int)(MT / GBM)), 256, 0, stream>>>(
          h0, W1, b1, nullptr, ff1, (int)MT, 2 * Dz, Dz);

  // 3) h0 <- ff1 @ W2 + b2 + h0     M=32768, N=512, K=1024 (residual fused)
  gemm_wmma_kernel<false, true>
      <<<dim3(Dz / GBN, (int)(MT / GBM)), 256, 0, stream>>>(
          ff1, W2, b2, h0, h0, (int)MT, Dz, 2 * Dz);

  // 4) h = LN(h0) in place; s_lo/s_hi gate row-dots fused
  ln_rowdot_kernel<<<(int)MT, 256, 0, stream>>>(h0, ln_g, ln_b, gate_W, s_lo,
                                                s_hi);

  // 5) gate = sigmoid(s_lo + mean_window(s_hi) + gate_b)
  gate_kernel<<<(int)((MT + 255) / 256), 256, 0, stream>>>(s_lo, s_hi, gate_b,
                                                           gate);

  // 6) exact top-256 indices per batch (rank-addressed, deterministic)
  topk_kernel<<<Bz, 1024, 0, stream>>>(gate, mem_idx);

  // 7) q = h[:, -1] @ q_W + q_b
  qproj_kernel<<<Bz, 256, 0, stream>>>(h0, q_W, q_b, q);

  // 8) softmax attention over memory slots -> ctx
  readhead_kernel<<<Bz, 256, 0, stream>>>(h0, mem_idx, q, ctx);

  // 9) out = ctx @ out_W + out_b
  outproj_kernel<<<(Vz + 255) / 256, 256, 0, stream>>>(ctx, out_W, out_b,
                                                       (float*)d_out);
}